// S2Network_57535381897392
// MI455X (gfx1250) — compile-verified
//
#include <hip/hip_runtime.h>
#include <cmath>
#include <complex>
#include <vector>
#include <algorithm>
#include <cstring>

// ---------------------------------------------------------------------------
// S2Network for MI455X (gfx1250): fused WMMA-f32 S2-activation + CG tensor ops
// ---------------------------------------------------------------------------

#define LMAXV   3
#define NSAMP   512
#define RESG    80
#define NGRID   (RESG*RESG)       // 6400
#define NDIM    16                // (LMAX+1)^2
#define MULH    16
#define NROWS   (NSAMP*MULH)      // 8192 rows in the (n,mul) flattened space

typedef float v2f __attribute__((ext_vector_type(2)));
typedef float v8f __attribute__((ext_vector_type(8)));

struct SegGPU { int l1,l2,l3,mul1,mul2,sym,cgOff,b1Off,b2Off,pOff,nP; };
struct TsParams {
  int inStride;
  int mulIn[4];       // channels taken from input, per l
  int inOff[4];       // per-l element offset within a sample's input vector
  int chanStride[4];  // per-l channel stride in input
  int ldsOff[4];      // staged-block offsets in LDS
  int linear;         // prepend constant-1 channel to l=0 block
};

// ============================ device kernels ===============================

// Tensor-square layer for one output l3: stage blocks -> CG paths -> weights.
__global__ __launch_bounds__(256)
void ts_kernel(const float* __restrict__ in, const float* __restrict__ cg,
               const SegGPU* __restrict__ segs, int nSegs,
               const float* __restrict__ W, int P, float invSqrtP,
               float* __restrict__ out, int outOff, int outNS, int outWS,
               int l3, TsParams prm)
{
  __shared__ float sB[320];     // staged input blocks (max 257 floats)
  __shared__ float sZ[9216];    // path values z[p][k] (max 1312*7 = 9184)
  const int n   = blockIdx.x;
  const int tid = threadIdx.x, nt = blockDim.x;

  if (prm.linear && tid == 0) sB[prm.ldsOff[0]] = 1.0f;
  for (int l = 0; l < 4; ++l) {
    const int dl = 2*l + 1;
    const int cnt = prm.mulIn[l] * dl;
    const int base = prm.ldsOff[l] + ((l == 0 && prm.linear) ? 1 : 0);
    for (int t = tid; t < cnt; t += nt) {
      const int m = t / dl, k = t % dl;
      sB[base + t] = in[(size_t)n*prm.inStride + prm.inOff[l] + m*prm.chanStride[l] + k];
    }
  }
  __syncthreads();

  const int d3 = 2*l3 + 1;
  for (int p = tid; p < P; p += nt) {
    int s = 0;
    while (s + 1 < nSegs && p >= segs[s+1].pOff) ++s;
    const SegGPU sg = segs[s];
    int loc = p - sg.pOff;
    const int d1 = 2*sg.l1 + 1, d2 = 2*sg.l2 + 1;
    int u, v;
    if (sg.sym) {           // decode np.triu_indices ordering
      int i = 0;
      while (loc >= sg.mul1 - i) { loc -= sg.mul1 - i; ++i; }
      u = i; v = i + loc;
    } else { u = loc / sg.mul2; v = loc % sg.mul2; }
    const float* b1 = &sB[sg.b1Off + u*d1];
    const float* b2 = &sB[sg.b2Off + v*d2];
    const float* C  = &cg[sg.cgOff];
    for (int k = 0; k < d3; ++k) {
      float acc = 0.f;
      for (int i = 0; i < d1; ++i) {
        const float bi = b1[i];
        for (int j = 0; j < d2; ++j) acc += C[(i*d2 + j)*d3 + k] * bi * b2[j];
      }
      sZ[p*d3 + k] = acc;
    }
  }
  __syncthreads();

  for (int t = tid; t < MULH*d3; t += nt) {
    const int w = t / d3, k = t % d3;
    float acc = 0.f;
    for (int p = 0; p < P; ++p) acc += W[w*P + p] * sZ[p*d3 + k];
    out[(size_t)n*outNS + w*outWS + outOff + k] = acc * invSqrtP;
  }
}

// Fused S2 activation:  out = (silu(C @ Y) * qw) @ Y^T,  all in fp32 WMMA.
// One wave per 16-row tile; loop over grid in 16-column chunks.
// LDS tile padded to 18 dwords/row so transpose reads are 8B-aligned b64
// and the stride-18 lane pattern stays bank-conflict free.
__global__ __launch_bounds__(256)
void s2act_kernel(const float* __restrict__ cin, const float* __restrict__ Y,
                  const float* __restrict__ QW, float* __restrict__ cout)
{
  __shared__ float sAct[8][16*18];          // per-wave D->A transpose tile
  const int lane = threadIdx.x & 31;
  const int wv   = threadIdx.x >> 5;
  const int tile = blockIdx.x * 8 + wv;     // 512 tiles of 16 rows
  const int m = lane & 15, h = lane >> 4;
  const int row = tile * 16 + m;

  // A (coeffs tile, M=row, K=d): VGPR0 = K {4c+2h}, VGPR1 = K {4c+2h+1}
  v2f a[4];
#pragma unroll
  for (int c = 0; c < 4; ++c)
    a[c] = *(const v2f*)&cin[row*NDIM + 4*c + 2*h];

  v8f acc = {};                             // back-projection accumulator
  float* sw = &sAct[wv][0];

  for (int gb = 0; gb < NGRID; gb += 16) {
    if (gb + 16 < NGRID) __builtin_prefetch(&Y[gb + 16], 0, 1);

    // forward B chunk: B[k][n] = Y[k][gb+n]  (row-strided, scalar loads)
    v2f b[4];
#pragma unroll
    for (int c = 0; c < 4; ++c) {
      b[c].x = Y[(4*c + 2*h)*NGRID + gb + m];
      b[c].y = Y[(4*c + 2*h + 1)*NGRID + gb + m];
    }
    v8f g = {};
#pragma unroll
    for (int c = 0; c < 4; ++c)
      g = __builtin_amdgcn_wmma_f32_16x16x4_f32(false, a[c], false, b[c],
                                                (short)0, g, false, false);

    // SiLU * quadrature weight via fast v_rcp (TRANS, co-executes with WMMA)
    const float q = QW[gb + m];
#pragma unroll
    for (int r = 0; r < 8; ++r) {
      const float v = g[r];
      const float sv = v * q * __builtin_amdgcn_rcpf(1.0f + __expf(-v));
      sw[(r + 8*h)*18 + m] = sv;
    }
    asm volatile("s_wait_dscnt 0" ::: "memory");

    // backward: A2[M][k]=act[M][gb+k] from LDS (b64); B2[k][n]=Y[n][gb+k] (b64)
    v2f a2[4], b2[4];
#pragma unroll
    for (int c = 0; c < 4; ++c) {
      a2[c] = *(const v2f*)&sw[m*18 + 4*c + 2*h];
      b2[c] = *(const v2f*)&Y[m*NGRID + gb + 4*c + 2*h];
    }
#pragma unroll
    for (int c = 0; c < 4; ++c)
      acc = __builtin_amdgcn_wmma_f32_16x16x4_f32(false, a2[c], false, b2[c],
                                                  (short)0, acc, false, false);
    asm volatile("s_wait_dscnt 0" ::: "memory");
  }

#pragma unroll
  for (int r = 0; r < 8; ++r)
    cout[(tile*16 + r + 8*h)*NDIM + m] = acc[r];
}

// ============================ host-side tables =============================

static double fct(int n) { double r = 1; for (int i = 2; i <= n; ++i) r *= i; return r; }

static double su2_cg(int j1,int m1,int j2,int m2,int j3,int m3) {
  if (m3 != m1 + m2) return 0.0;
  int vmin = std::max(std::max(-j1 + j2 + m3, -j1 + m1), 0);
  int vmax = std::min(std::min(j2 + j3 + m1, j3 - j1 + j2), j3 + m3);
  double c = std::sqrt((2*j3+1)*fct(j3+j1-j2)*fct(j3-j1+j2)*fct(j1+j2-j3)*fct(j3+m3)*fct(j3-m3)
             /(fct(j1+j2+j3+1)*fct(j1-m1)*fct(j1+m1)*fct(j2-m2)*fct(j2+m2)));
  double s = 0.0;
  for (int v = vmin; v <= vmax; ++v) {
    double sg = ((v + j2 + m2) % 2 == 0) ? 1.0 : -1.0;
    s += sg * fct(j2+j3+m1-v)*fct(j1-m1+v)
           / (fct(v)*fct(j3-j1+j2-v)*fct(j3+m3-v)*fct(v+j1-j2-m3));
  }
  return c * s;
}

typedef std::complex<double> cd;

static void qmat(int l, cd* q) {
  const int d = 2*l + 1;
  for (int i = 0; i < d*d; ++i) q[i] = cd(0,0);
  const double r = std::sqrt(0.5);
  for (int m = -l; m < 0; ++m) {
    q[(l+m)*d + (l - m)] = cd(r, 0);           // l+|m| = l-m for m<0
    q[(l+m)*d + (l + m)] = cd(0, -r);          // l-|m| = l+m for m<0
  }
  q[l*d + l] = cd(1, 0);
  for (int m = 1; m <= l; ++m) {
    double sg = (m % 2 == 0) ? 1.0 : -1.0;
    q[(l+m)*d + (l + m)] = cd(sg*r, 0);
    q[(l+m)*d + (l - m)] = cd(0, sg*r);
  }
  cd f(1,0); for (int i = 0; i < l; ++i) f *= cd(0,-1);
  for (int i = 0; i < d*d; ++i) q[i] *= f;
}

static void cg_table(int l1,int l2,int l3, float* out) {
  const int d1=2*l1+1, d2=2*l2+1, d3=2*l3+1;
  std::vector<cd> Cc(d1*d2*d3, cd(0,0));
  for (int m1=-l1; m1<=l1; ++m1) for (int m2=-l2; m2<=l2; ++m2) {
    int m3 = m1 + m2;
    if (std::abs(m3) <= l3)
      Cc[((l1+m1)*d2 + (l2+m2))*d3 + (l3+m3)] = su2_cg(l1,m1,l2,m2,l3,m3);
  }
  std::vector<cd> q1(d1*d1), q2(d2*d2), q3(d3*d3);
  qmat(l1,q1.data()); qmat(l2,q2.data()); qmat(l3,q3.data());
  std::vector<cd> C(d1*d2*d3, cd(0,0));
  for (int j=0;j<d1;++j) for (int li=0;li<d2;++li) for (int m=0;m<d3;++m) {
    cd s(0,0);
    for (int i=0;i<d1;++i) for (int k=0;k<d2;++k) for (int n=0;n<d3;++n)
      s += q1[i*d1+j]*q2[k*d2+li]*std::conj(q3[n*d3+m])*Cc[(i*d2+k)*d3+n];
    C[(j*d2+li)*d3+m] = s;
  }
  double nr=0, nC=0;
  for (auto& v : C) { nr += v.real()*v.real(); nC += std::norm(v); }
  bool useImag = std::sqrt(nr) < 1e-6*std::max(std::sqrt(nC), 1e-30);
  double nn = 0;
  for (int t = 0; t < d1*d2*d3; ++t) {
    double x = useImag ? C[t].imag() : C[t].real();
    out[t] = (float)x; nn += x*x;
  }
  nn = std::sqrt(nn);
  if (nn > 0) for (int t = 0; t < d1*d2*d3; ++t) out[t] /= (float)nn;
}

static void s2_grid(float* Y, float* QW) {
  const double PI = 3.14159265358979323846;
  double ct[RESG], st[RESG], alphas[RESG];
  for (int b = 0; b < RESG; ++b) {
    double beta = (b + 0.5)/RESG*PI; ct[b] = std::cos(beta); st[b] = std::sin(beta);
  }
  for (int a = 0; a < RESG; ++a) alphas[a] = (double)a/RESG*2.0*PI;
  static double Pl[4][4][RESG];
  for (int m = 0; m <= LMAXV; ++m) {
    double dfact = 1; for (int k = 1; k < 2*m; k += 2) dfact *= k;
    double sg = (m % 2) ? -1.0 : 1.0;
    for (int b = 0; b < RESG; ++b) Pl[m][m][b] = sg*dfact*std::pow(st[b], m);
    if (m < LMAXV) for (int b = 0; b < RESG; ++b) Pl[m+1][m][b] = ct[b]*(2*m+1)*Pl[m][m][b];
    for (int l = m+2; l <= LMAXV; ++l)
      for (int b = 0; b < RESG; ++b)
        Pl[l][m][b] = ((2*l-1)*ct[b]*Pl[l-1][m][b] - (l+m-1)*Pl[l-2][m][b])/(l-m);
  }
  int i = 0;
  for (int l = 0; l <= LMAXV; ++l) for (int m = -l; m <= l; ++m) {
    int am = std::abs(m);
    double k = std::sqrt((2*l+1)/(4*PI)*fct(l-am)/fct(l+am));
    for (int b = 0; b < RESG; ++b) for (int a = 0; a < RESG; ++a) {
      double v;
      if (m == 0)      v = k*Pl[l][0][b];
      else if (m > 0)  v = std::sqrt(2.0)*k*Pl[l][m][b]*std::cos(m*alphas[a]);
      else             v = std::sqrt(2.0)*k*Pl[l][am][b]*std::sin(am*alphas[a]);
      Y[i*NGRID + b*RESG + a] = (float)v;
    }
    ++i;
  }
  for (int b = 0; b < RESG; ++b) for (int a = 0; a < RESG; ++a)
    QW[b*RESG + a] = (float)(st[b]*(PI/RESG)*(2.0*PI/RESG));
}

static int gen_segs(const int mulEff[4], const int ldsOff[4], int lmaxOut,
                    SegGPU* segs, int segBase,
                    int start[4], int cnt[4], int Pout[4],
                    int cgOffTab[4][4][4])
{
  for (int l3 = 0; l3 <= lmaxOut; ++l3) {
    start[l3] = segBase; int pOff = 0;
    for (int l1 = 0; l1 <= 3; ++l1) {
      if (!mulEff[l1]) continue;
      for (int l2 = l1; l2 <= 3; ++l2) {
        if (!mulEff[l2]) continue;
        if (l3 < std::abs(l1-l2) || l3 > std::min(l1+l2, LMAXV)) continue;
        SegGPU s;
        s.l1 = l1; s.l2 = l2; s.l3 = l3;
        s.sym = (l1 == l2);
        s.mul1 = mulEff[l1]; s.mul2 = mulEff[l2];
        s.b1Off = ldsOff[l1]; s.b2Off = ldsOff[l2];
        s.cgOff = cgOffTab[l1][l2][l3];
        s.pOff = pOff;
        s.nP = s.sym ? mulEff[l1]*(mulEff[l1]+1)/2 : mulEff[l1]*mulEff[l2];
        pOff += s.nP;
        segs[segBase++] = s;
      }
    }
    cnt[l3] = segBase - start[l3];
    Pout[l3] = pOff;
  }
  return segBase;
}

// =============================== launcher ==================================

extern "C" void kernel_launch(void* const* d_in, const int* in_sizes, int n_in,
                              void* d_out, int out_size, void* d_ws, size_t ws_size,
                              hipStream_t stream)
{
  (void)in_sizes; (void)n_in; (void)out_size; (void)ws_size;

  // ---- build deterministic host tables (stable static storage) ----
  static float  hY[NDIM*NGRID];
  static float  hQW[NGRID];
  static float  hCG[4096];
  static SegGPU hSegs[96];
  static int    cgOffTab[4][4][4];

  s2_grid(hY, hQW);
  int cgTotal = 0;
  for (int l1 = 0; l1 <= 3; ++l1)
    for (int l2 = l1; l2 <= 3; ++l2)
      for (int l3 = std::abs(l1-l2); l3 <= std::min(l1+l2, LMAXV); ++l3) {
        cgOffTab[l1][l2][l3] = cgTotal;
        cg_table(l1, l2, l3, hCG + cgTotal);
        cgTotal += (2*l1+1)*(2*l2+1)*(2*l3+1);
      }

  // layer configs
  TsParams prmA = {};   // first layer, input = x (512,36), linear
  prmA.inStride = 36; prmA.linear = 1;
  { int mi[4] = {4,4,4,0}, io[4] = {0,4,16,0}, cs[4] = {1,3,5,0};
    for (int l=0;l<4;++l){ prmA.mulIn[l]=mi[l]; prmA.inOff[l]=io[l]; prmA.chanStride[l]=cs[l]; } }
  int mulA[4] = {5,4,4,0};
  { int off=0; for (int l=0;l<4;++l){ prmA.ldsOff[l]=off; off += mulA[l]*(2*l+1); } }

  TsParams prmB = {};   // hidden layers, input = coeffs (512,16,16), linear
  prmB.inStride = 256; prmB.linear = 1;
  for (int l=0;l<4;++l){ prmB.mulIn[l]=16; prmB.inOff[l]=l*l; prmB.chanStride[l]=16; }
  int mulB[4] = {17,16,16,16};
  { int off=0; for (int l=0;l<4;++l){ prmB.ldsOff[l]=off; off += mulB[l]*(2*l+1); } }

  TsParams prmC = prmB; prmC.linear = 0;   // final layer, no linear
  int mulC[4] = {16,16,16,16};
  { int off=0; for (int l=0;l<4;++l){ prmC.ldsOff[l]=off; off += mulC[l]*(2*l+1); } }

  static int stA[4], cnA[4], PA[4], stB[4], cnB[4], PB[4], stC[4], cnC[4], PC[4];
  int nSegTot = 0;
  nSegTot = gen_segs(mulA, prmA.ldsOff, 3, hSegs, nSegTot, stA, cnA, PA, cgOffTab);
  nSegTot = gen_segs(mulB, prmB.ldsOff, 3, hSegs, nSegTot, stB, cnB, PB, cgOffTab);
  nSegTot = gen_segs(mulC, prmC.ldsOff, 0, hSegs, nSegTot, stC, cnC, PC, cgOffTab);

  // ---- carve workspace ----
  float*  dY   = (float*)d_ws;
  float*  dQW  = dY + NDIM*NGRID;
  float*  dCG  = dQW + NGRID;
  float*  buf0 = dCG + 4096;
  float*  buf1 = buf0 + (size_t)NSAMP*256;
  SegGPU* dSegs = (SegGPU*)(buf1 + (size_t)NSAMP*256);

  hipMemcpyAsync(dY,   hY,   sizeof(hY),                hipMemcpyHostToDevice, stream);
  hipMemcpyAsync(dQW,  hQW,  sizeof(hQW),               hipMemcpyHostToDevice, stream);
  hipMemcpyAsync(dCG,  hCG,  (size_t)cgTotal*4,         hipMemcpyHostToDevice, stream);
  hipMemcpyAsync(dSegs,hSegs,(size_t)nSegTot*sizeof(SegGPU), hipMemcpyHostToDevice, stream);

  const float* x = (const float*)d_in[0];
  float* outp = (float*)d_out;

  // layer 0: x -> buf0, then fused S2 act -> buf1
  for (int l3 = 0; l3 < 4; ++l3)
    ts_kernel<<<NSAMP, 256, 0, stream>>>(x, dCG, dSegs + stA[l3], cnA[l3],
        (const float*)d_in[1 + l3], PA[l3], 1.0f/std::sqrt((float)PA[l3]),
        buf0, l3*l3, 256, 16, l3, prmA);
  s2act_kernel<<<NROWS/16/8, 256, 0, stream>>>(buf0, dY, dQW, buf1);

  // layers 1,2: hidden ts + act (act always reads buf0, writes buf1)
  for (int layer = 1; layer <= 2; ++layer) {
    for (int l3 = 0; l3 < 4; ++l3)
      ts_kernel<<<NSAMP, 256, 0, stream>>>(buf1, dCG, dSegs + stB[l3], cnB[l3],
          (const float*)d_in[1 + 4*layer + l3], PB[l3], 1.0f/std::sqrt((float)PB[l3]),
          buf0, l3*l3, 256, 16, l3, prmB);
    s2act_kernel<<<NROWS/16/8, 256, 0, stream>>>(buf0, dY, dQW, buf1);
  }

  // final scalar tensor product: buf1 -> d_out (512,16)
  ts_kernel<<<NSAMP, 256, 0, stream>>>(buf1, dCG, dSegs + stC[0], cnC[0],
      (const float*)d_in[13], PC[0], 1.0f/std::sqrt((float)PC[0]),
      outp, 0, 16, 1, 0, prmC);
}